// ImprovedClusteringLoss_68771016343846
// MI455X (gfx1250) — compile-verified
//
#include <hip/hip_runtime.h>
#include <hip/hip_bf16.h>

#define DIM        128
#define C_CLASSES  100
#define MARGIN     2.0f
#define EPS_V      1e-6f
#define CT         7           // ceil(100/16) tiles per side
#define CPAD       (CT * 16)   // 112 padded rows
#define LDS_STRIDE 130         // 130 % 64 = 2 -> conflict-free row access, 8B aligned

typedef __attribute__((ext_vector_type(2))) float v2f;
typedef __attribute__((ext_vector_type(8))) float v8f;

// ---------------------------------------------------------------------------
// Kernel 0: zero the workspace accumulators (d_ws not re-poisoned by harness)
// ---------------------------------------------------------------------------
__global__ void icl_init_kernel(float* ws) {
    if (threadIdx.x < 2) ws[threadIdx.x] = 0.0f;
}

// ---------------------------------------------------------------------------
// Kernel 1: center loss partial sums. One wave32 per feature row per iter.
// Lane l loads features[row*128 + l*4 .. +3] as float4 (global_load_b128).
// Memory-bound: 256 MB of features streamed once (~11 us at 23.3 TB/s).
// ---------------------------------------------------------------------------
__global__ __launch_bounds__(256) void icl_center_kernel(
    const float* __restrict__ features,
    const int*   __restrict__ labels,
    const float* __restrict__ centers,
    float* __restrict__ ws,
    int n_rows)
{
    const int lane   = threadIdx.x & 31;
    const int waveId = (blockIdx.x * blockDim.x + threadIdx.x) >> 5;
    const int nWaves = (gridDim.x * blockDim.x) >> 5;

    float acc = 0.0f;
    for (int row = waveId; row < n_rows; row += nWaves) {
        const int lbl = labels[row];
        const float4 fv = ((const float4*)(features + (size_t)row * DIM))[lane];
        const float4 cv = ((const float4*)(centers  + (size_t)lbl * DIM))[lane];
        const float dx = fv.x - cv.x, dy = fv.y - cv.y;
        const float dz = fv.z - cv.z, dw = fv.w - cv.w;
        acc = fmaf(dx, dx, acc);
        acc = fmaf(dy, dy, acc);
        acc = fmaf(dz, dz, acc);
        acc = fmaf(dw, dw, acc);
    }

    // wave32 reduction
    for (int off = 16; off > 0; off >>= 1)
        acc += __shfl_down(acc, off, 32);

    __shared__ float sAcc[8];
    const int w = threadIdx.x >> 5;
    if (lane == 0) sAcc[w] = acc;
    __syncthreads();
    if (threadIdx.x == 0) {
        float s = 0.0f;
        #pragma unroll
        for (int i = 0; i < 8; ++i) s += sAcc[i];
        atomicAdd(&ws[0], s);
    }
}

// ---------------------------------------------------------------------------
// Kernel 2: separation loss via WMMA Gram matrix.
//   G = C * C^T with V_WMMA_F32_16X16X4_F32 (f32 data -> f32 WMMA path)
//   dist^2(i,j) = n_i + n_j - 2 G_ij + 2*eps*(s_i - s_j) + D*eps^2
// Centers are staged zero-padded (112 rows) into LDS so the WMMA K-loop is a
// branch-free stream of ds_load_b64 + v_wmma with EXEC all-1s throughout.
// ---------------------------------------------------------------------------
__global__ __launch_bounds__(256) void icl_sep_kernel(
    const float* __restrict__ centers,
    float* __restrict__ ws)
{
    __shared__ float ldsC[CPAD * LDS_STRIDE];  // 58,240 B, zero-padded centers
    __shared__ float sN[CPAD];                 // squared norms
    __shared__ float sS[CPAD];                 // row sums
    __shared__ float sAcc[8];

    const int t = threadIdx.x;

    // Phase 0: cooperative zero-padded stage of centers into LDS.
    // 112*128 = 14336 elements / 256 threads = 56 each; DIM=128 -> shifts.
    for (int idx = t; idx < CPAD * DIM; idx += 256) {
        const int r = idx >> 7;        // / DIM
        const int d = idx & (DIM - 1); // % DIM
        const float v = (r < C_CLASSES) ? centers[(size_t)r * DIM + d] : 0.0f;
        ldsC[r * LDS_STRIDE + d] = v;
    }
    __syncthreads();

    // Phase 1: per-row norm and sum from LDS (padded rows give 0 naturally).
    if (t < CPAD) {
        float nv = 0.0f, sv = 0.0f;
        const float* row = &ldsC[t * LDS_STRIDE];
        #pragma unroll 4
        for (int d = 0; d < DIM; ++d) {
            const float v = row[d];
            nv = fmaf(v, v, nv);
            sv += v;
        }
        sN[t] = nv;
        sS[t] = sv;
    }
    __syncthreads();

    const int wave = t >> 5;
    const int lane = t & 31;
    float acc = 0.0f;

    // Phase 2: WMMA Gram tiles. A-frag layout (16x4 f32): lanes 0-15 hold
    // K={0,1}, lanes 16-31 hold K={2,3}; m = lane&15. B mirrors with n=lane&15.
    const int m     = lane & 15;
    const int khalf = lane >> 4;

    for (int tile = wave; tile < CT * CT; tile += 8) {
        const int I = tile / CT;
        const int J = tile % CT;

        // Padded rows are zero in LDS -> no bounds checks, EXEC stays all-1s.
        const float* baseA = &ldsC[(I * 16 + m) * LDS_STRIDE + khalf * 2];
        const float* baseB = &ldsC[(J * 16 + m) * LDS_STRIDE + khalf * 2];

        v8f c = {};
        #pragma unroll 8
        for (int k = 0; k < DIM; k += 4) {
            const v2f a = *(const v2f*)(baseA + k);   // ds_load_b64
            const v2f b = *(const v2f*)(baseB + k);   // ds_load_b64
            // (neg_a, A, neg_b, B, c_mod, C, reuse_a, reuse_b)
            c = __builtin_amdgcn_wmma_f32_16x16x4_f32(
                    false, a, false, b, (short)0, c, false, false);
        }

        // C/D layout: VGPR v -> lanes 0-15: M=v, N=lane; lanes 16-31: M=v+8.
        const int Nn    = lane & 15;
        const int Mbase = (lane >> 4) * 8;
        const int j     = J * 16 + Nn;
        #pragma unroll
        for (int v = 0; v < 8; ++v) {
            const int i = I * 16 + Mbase + v;
            if (i < C_CLASSES && j < C_CLASSES && i < j) {
                const float g = c[v];
                float d2 = sN[i] + sN[j] - 2.0f * g
                         + 2.0f * EPS_V * (sS[i] - sS[j])
                         + (float)DIM * EPS_V * EPS_V;
                d2 = fmaxf(d2, 0.0f);
                const float h = MARGIN - __builtin_sqrtf(d2);
                if (h > 0.0f) acc += h;
            }
        }
    }

    for (int off = 16; off > 0; off >>= 1)
        acc += __shfl_down(acc, off, 32);
    if (lane == 0) sAcc[wave] = acc;
    __syncthreads();
    if (t == 0) {
        float s = 0.0f;
        #pragma unroll
        for (int w = 0; w < 8; ++w) s += sAcc[w];
        ws[1] = s;
    }
}

// ---------------------------------------------------------------------------
// Kernel 3: finalize -> d_out = {center_sum/(N*D), sep_sum/num_pairs}
// ---------------------------------------------------------------------------
__global__ void icl_final_kernel(const float* __restrict__ ws,
                                 float* __restrict__ out, int n_rows)
{
    if (threadIdx.x == 0) {
        out[0] = ws[0] / ((float)n_rows * (float)DIM);
        out[1] = ws[1] / (float)((C_CLASSES * (C_CLASSES - 1)) / 2);
    }
}

extern "C" void kernel_launch(void* const* d_in, const int* in_sizes, int n_in,
                              void* d_out, int out_size, void* d_ws, size_t ws_size,
                              hipStream_t stream) {
    const float* features = (const float*)d_in[0];
    const int*   labels   = (const int*)d_in[1];
    const float* centers  = (const float*)d_in[2];
    float* out = (float*)d_out;
    float* ws  = (float*)d_ws;

    const int n_rows = in_sizes[0] / DIM;   // 524288

    icl_init_kernel<<<1, 32, 0, stream>>>(ws);

    // Memory-bound streaming pass: 2048 blocks x 8 waves, 32 rows/wave,
    // b128 loads per lane to keep 23.3 TB/s HBM fed.
    icl_center_kernel<<<2048, 256, 0, stream>>>(features, labels, centers, ws, n_rows);

    // One workgroup handles the 100x100 Gram + hinge via f32 WMMA from LDS.
    icl_sep_kernel<<<1, 256, 0, stream>>>(centers, ws);

    icl_final_kernel<<<1, 32, 0, stream>>>(ws, out, n_rows);
}